// ShallowGCNConvNet_16561393893735
// MI455X (gfx1250) — compile-verified
//
#include <hip/hip_runtime.h>
#include <hip/hip_bf16.h>
#include <stdint.h>

// ---------------- problem constants ----------------
#define NN 50000
#define EE 300000
#define NP 50048          // N padded to multiple of 128 (GEMM M-tile)
#define BN_EPS 1e-5f

typedef __attribute__((ext_vector_type(2))) float v2f;
typedef __attribute__((ext_vector_type(8))) float v8f;

// ---------------- utility kernels ----------------
__global__ void fill_zero(float* __restrict__ p, long n) {
  long i = (long)blockIdx.x * blockDim.x + threadIdx.x;
  long stride = (long)gridDim.x * blockDim.x;
  for (; i < n; i += stride) p[i] = 0.0f;
}

__global__ void pack_x(const float* __restrict__ x, float* __restrict__ xp) {
  long n = (long)NP * 128;
  long i = (long)blockIdx.x * blockDim.x + threadIdx.x;
  long stride = (long)gridDim.x * blockDim.x;
  for (; i < n; i += stride) {
    long row = i >> 7;
    xp[i] = (row < NN) ? x[i] : 0.0f;
  }
}

__global__ void pack_w(const float* __restrict__ W, float* __restrict__ Wp,
                       int K, int Fo, int Kp, int Fop) {
  long n = (long)Kp * Fop;
  long i = (long)blockIdx.x * blockDim.x + threadIdx.x;
  long stride = (long)gridDim.x * blockDim.x;
  for (; i < n; i += stride) {
    int k = (int)(i / Fop);       // Fop is a power of two
    int f = (int)(i - (long)k * Fop);
    Wp[i] = (k < K && f < Fo) ? W[(long)k * Fo + f] : 0.0f;
  }
}

__global__ void deg_kernel(const int* __restrict__ dst, float* __restrict__ deg) {
  int e = blockIdx.x * blockDim.x + threadIdx.x;
  if (e < EE) atomicAdd(&deg[dst[e]], 1.0f);
}

__global__ void dis_kernel(const float* __restrict__ deg, float* __restrict__ dis) {
  int i = blockIdx.x * blockDim.x + threadIdx.x;
  if (i < NN) dis[i] = rsqrtf(deg[i] + 1.0f);
}

// ---------------- WMMA fp32 GEMM: C[NP,Fop] = A[NP,Kp] * B[Kp,Fop] ----------------
// Block tile 128x64, K-chunk 32, 8 waves; each wave owns a 16x64 strip (4 accs).
// Double-buffered LDS filled by GLOBAL_LOAD_ASYNC_TO_LDS_B128 (ASYNCcnt),
// pipelined against the WMMA inner loop.
#define TM 128
#define TN 64
#define TK 32
#define AS_STRIDE 36   // bank-conflict-free A fragment reads (gcd(36,64)=4 spread)
#define BS_STRIDE 72   // lanes0-15 / lanes16-31 rows land on disjoint bank ranges

__global__ __launch_bounds__(256) void gemm_wmma(const float* __restrict__ A,
                                                 const float* __restrict__ B,
                                                 float* __restrict__ C,
                                                 int Kp, int Fop) {
  __shared__ float As[2][TM * AS_STRIDE];   // 2 x 18.0 KB
  __shared__ float Bs[2][TK * BS_STRIDE];   // 2 x  9.0 KB   (total ~55.3 KB)

  const int tid  = threadIdx.x;
  const int wave = tid >> 5;
  const int lane = tid & 31;
  const int half = lane >> 4;
  const int l16  = lane & 15;

  const long rowBase = (long)blockIdx.x * TM;
  const int  colBase = blockIdx.y * TN;

  v8f acc0 = {}, acc1 = {}, acc2 = {}, acc3 = {};

  // Issue async DMA of one K-chunk (A: 128x32 floats, B: 32x64 floats) into LDS buf.
  auto issue_chunk = [&](int kc, int buf) {
    #pragma unroll
    for (int p = 0; p < 4; ++p) {              // A tile: 4 x float4 per thread
      int linear = p * 256 + tid;
      int r  = linear >> 3;                    // 8 float4 per row of 32
      int c4 = (linear & 7) << 2;
      unsigned lds = (unsigned)(uintptr_t)&As[buf][r * AS_STRIDE + c4];
      const float* g = A + (rowBase + r) * (long)Kp + kc + c4;
      asm volatile("global_load_async_to_lds_b128 %0, %1, off"
                   :: "v"(lds), "v"(g) : "memory");
    }
    #pragma unroll
    for (int p = 0; p < 2; ++p) {              // B tile: 2 x float4 per thread
      int linear = p * 256 + tid;
      int r  = linear >> 4;                    // 16 float4 per row of 64
      int c4 = (linear & 15) << 2;
      unsigned lds = (unsigned)(uintptr_t)&Bs[buf][r * BS_STRIDE + c4];
      const float* g = B + (long)(kc + r) * Fop + colBase + c4;
      asm volatile("global_load_async_to_lds_b128 %0, %1, off"
                   :: "v"(lds), "v"(g) : "memory");
    }
  };

  issue_chunk(0, 0);

  const int nChunks = Kp / TK;
  for (int ch = 0; ch < nChunks; ++ch) {
    const int buf = ch & 1;
    // my wave's async fills done; barrier makes it workgroup-wide
    asm volatile("s_wait_asynccnt 0x0" ::: "memory");
    __syncthreads();
    if (ch + 1 < nChunks) issue_chunk((ch + 1) * TK, buf ^ 1);  // overlap DMA w/ math

    const float* as = &As[buf][0];
    const float* bs = &Bs[buf][0];
    const int arow = wave * 16 + l16;
    #pragma unroll
    for (int kk = 0; kk < TK; kk += 4) {
      // A fragment 16x4: lane = M; VGPR pair holds K, half selects K+2 (contiguous b64)
      const int ka = kk + half * 2;
      v2f a = *(const v2f*)(as + arow * AS_STRIDE + ka);
      // B fragment 4x16: lane = N; VGPR = K within pair, half selects K+2
      const int kb = kk + half * 2;
      v2f b0, b1, b2, b3;
      b0.x = bs[kb * BS_STRIDE +  0 + l16]; b0.y = bs[(kb + 1) * BS_STRIDE +  0 + l16];
      b1.x = bs[kb * BS_STRIDE + 16 + l16]; b1.y = bs[(kb + 1) * BS_STRIDE + 16 + l16];
      b2.x = bs[kb * BS_STRIDE + 32 + l16]; b2.y = bs[(kb + 1) * BS_STRIDE + 32 + l16];
      b3.x = bs[kb * BS_STRIDE + 48 + l16]; b3.y = bs[(kb + 1) * BS_STRIDE + 48 + l16];

      acc0 = __builtin_amdgcn_wmma_f32_16x16x4_f32(false, a, false, b0, (short)0, acc0, false, false);
      acc1 = __builtin_amdgcn_wmma_f32_16x16x4_f32(false, a, false, b1, (short)0, acc1, false, false);
      acc2 = __builtin_amdgcn_wmma_f32_16x16x4_f32(false, a, false, b2, (short)0, acc2, false, false);
      acc3 = __builtin_amdgcn_wmma_f32_16x16x4_f32(false, a, false, b3, (short)0, acc3, false, false);
    }
  }

  // Store: C/D layout — VGPR j: lanes0-15 row j, lanes16-31 row j+8
  const long crow0 = rowBase + wave * 16 + half * 8;
  #pragma unroll
  for (int j = 0; j < 8; ++j) {
    long r = (crow0 + j) * (long)Fop + colBase + l16;
    C[r +  0] = acc0[j];
    C[r + 16] = acc1[j];
    C[r + 32] = acc2[j];
    C[r + 48] = acc3[j];
  }
}

// ---------------- aggregation ----------------
// O[i][f] = bias[f] + self_norm(i) * H[i][f]   (pad cols -> 0, pad rows -> bias)
__global__ void agg_init(const float* __restrict__ H, const float* __restrict__ bias,
                         const float* __restrict__ dis, float* __restrict__ O,
                         int Fo, int Fop) {
  long n = (long)NP * Fop;
  long idx = (long)blockIdx.x * blockDim.x + threadIdx.x;
  long stride = (long)gridDim.x * blockDim.x;
  for (; idx < n; idx += stride) {
    long i = idx / Fop;                 // Fop is power of two
    int  f = (int)(idx - i * Fop);
    float v = 0.0f;
    if (f < Fo) {
      v = bias[f];
      if (i < NN) {
        float d = dis[i];
        v += d * d * H[idx];
      }
    }
    O[idx] = v;
  }
}

// O[dst] += dis[src]*dis[dst] * H[src]; 4 edges per block, 64 lanes per edge
__global__ __launch_bounds__(256) void agg_edges(const int* __restrict__ src,
                                                 const int* __restrict__ dst,
                                                 const float* __restrict__ dis,
                                                 const float* __restrict__ H,
                                                 float* __restrict__ O,
                                                 int Fo, int Fop) {
  int e = blockIdx.x * 4 + (threadIdx.x >> 6);
  if (e >= EE) return;
  int s = src[e], d = dst[e];
  float w = dis[s] * dis[d];
  const float* hs = H + (long)s * Fop;
  float* od = O + (long)d * Fop;
  for (int f = threadIdx.x & 63; f < Fo; f += 64)
    atomicAdd(&od[f], w * hs[f]);
}

// ---------------- ELU + BatchNorm ----------------
#define BN_ROWS 2048
__global__ __launch_bounds__(256) void elu_stats(float* __restrict__ O,
                                                 float* __restrict__ sum,
                                                 float* __restrict__ sumsq,
                                                 int Fo, int Fop) {
  __shared__ float s1[4][64];
  __shared__ float s2[4][64];
  int tx = threadIdx.x & 63;
  int g  = threadIdx.x >> 6;
  int f  = blockIdx.x * 64 + tx;
  long r0 = (long)blockIdx.y * BN_ROWS;
  long r1 = r0 + BN_ROWS;
  if (r1 > (long)NN) r1 = NN;
  float a = 0.0f, b = 0.0f;
  if (f < Fo) {
    for (long i = r0 + g; i < r1; i += 4) {
      long idx = i * Fop + f;
      float v = O[idx];
      v = (v > 0.0f) ? v : expm1f(v);   // ELU, alpha = 1
      O[idx] = v;
      a += v;
      b += v * v;
    }
  }
  s1[g][tx] = a;
  s2[g][tx] = b;
  __syncthreads();
  if (g == 0 && f < Fo) {
    a = s1[0][tx] + s1[1][tx] + s1[2][tx] + s1[3][tx];
    b = s2[0][tx] + s2[1][tx] + s2[2][tx] + s2[3][tx];
    atomicAdd(&sum[f], a);
    atomicAdd(&sumsq[f], b);
  }
}

__global__ void bn_final(const float* __restrict__ sum, const float* __restrict__ sumsq,
                         const float* __restrict__ g, const float* __restrict__ be,
                         float* __restrict__ scale, float* __restrict__ shift, int Fo) {
  int f = blockIdx.x * blockDim.x + threadIdx.x;
  if (f >= Fo) return;
  float m  = sum[f] * (1.0f / (float)NN);
  float v  = sumsq[f] * (1.0f / (float)NN) - m * m;
  float sc = g[f] * rsqrtf(v + BN_EPS);
  scale[f] = sc;
  shift[f] = be[f] - m * sc;
}

__global__ void bn_apply(float* __restrict__ O, const float* __restrict__ scale,
                         const float* __restrict__ shift, int Fo, int Fop) {
  long n = (long)NP * Fo;
  long idx = (long)blockIdx.x * blockDim.x + threadIdx.x;
  long stride = (long)gridDim.x * blockDim.x;
  for (; idx < n; idx += stride) {
    long i = idx / Fo;
    int  f = (int)(idx - i * Fo);
    long p = i * Fop + f;
    O[p] = O[p] * scale[f] + shift[f];
  }
}

__global__ void final_copy(const float* __restrict__ O, float* __restrict__ out) {
  long n = (long)NN * 50;
  long idx = (long)blockIdx.x * blockDim.x + threadIdx.x;
  long stride = (long)gridDim.x * blockDim.x;
  for (; idx < n; idx += stride) {
    long i = idx / 50;
    int  f = (int)(idx - i * 50);
    out[idx] = O[i * 64 + f];
  }
}

// ---------------- host orchestration ----------------
extern "C" void kernel_launch(void* const* d_in, const int* in_sizes, int n_in,
                              void* d_out, int out_size, void* d_ws, size_t ws_size,
                              hipStream_t stream) {
  const float* x   = (const float*)d_in[0];
  const int*   src = (const int*)d_in[1];
  const int*   dst = (const int*)d_in[2];

  static const int KP[4]  = {128, 512, 1024, 512};   // padded K
  static const int FOP[4] = {512, 1024, 512, 64};    // padded out features
  static const int KR[4]  = {128, 500, 1000, 500};   // real K
  static const int FR[4]  = {500, 1000, 500, 50};    // real out features

  float* ws = (float*)d_ws;
  float* buf0  = ws;
  float* buf1  = buf0 + (long)NP * 1024;
  float* buf2  = buf1 + (long)NP * 1024;
  float* Wp    = buf2 + (long)NP * 1024;   // 512*1024 floats max
  float* dis   = Wp + (long)512 * 1024;
  float* deg   = dis + NN;
  float* sum   = deg + NN;
  float* sumsq = sum + 1024;
  float* scl   = sumsq + 1024;
  float* shf   = scl + 1024;

  const int T = 256;
  const int GS = 2048;  // grid-stride launch width

  // degrees -> dis = rsqrt(deg+1)
  fill_zero<<<GS, T, 0, stream>>>(deg, NN);
  deg_kernel<<<(EE + T - 1) / T, T, 0, stream>>>(dst, deg);
  dis_kernel<<<(NN + T - 1) / T, T, 0, stream>>>(deg, dis);

  // padded input
  pack_x<<<GS, T, 0, stream>>>(x, buf0);

  float* X = buf0;
  float* H = buf1;
  float* O = buf2;

  for (int l = 0; l < 4; ++l) {
    const float* W  = (const float*)d_in[3 + 4 * l];
    const float* b  = (const float*)d_in[4 + 4 * l];
    const int Kp = KP[l], Fop = FOP[l], K = KR[l], Fo = FR[l];

    pack_w<<<GS, T, 0, stream>>>(W, Wp, K, Fo, Kp, Fop);

    dim3 gg(NP / TM, Fop / TN);
    gemm_wmma<<<gg, T, 0, stream>>>(X, Wp, H, Kp, Fop);

    agg_init<<<GS, T, 0, stream>>>(H, b, dis, O, Fo, Fop);
    agg_edges<<<(EE + 3) / 4, T, 0, stream>>>(src, dst, dis, H, O, Fo, Fop);

    if (l < 3) {
      const float* g  = (const float*)d_in[5 + 4 * l];
      const float* be = (const float*)d_in[6 + 4 * l];
      fill_zero<<<8, T, 0, stream>>>(sum, 1024);
      fill_zero<<<8, T, 0, stream>>>(sumsq, 1024);
      dim3 gs((Fo + 63) / 64, (NN + BN_ROWS - 1) / BN_ROWS);
      elu_stats<<<gs, T, 0, stream>>>(O, sum, sumsq, Fo, Fop);
      bn_final<<<(Fo + T - 1) / T, T, 0, stream>>>(sum, sumsq, g, be, scl, shf, Fo);
      bn_apply<<<GS, T, 0, stream>>>(O, scl, shf, Fo, Fop);
    }

    // rotate buffers: next X is this layer's O
    float* tmpX = X;
    X = O;
    O = H;
    H = tmpX;
  }

  // X now points at the last layer's aggregated output (Fop = 64, Fo = 50)
  final_copy<<<GS, T, 0, stream>>>(X, (float*)d_out);
}